// QAPLoss_31464930410733
// MI455X (gfx1250) — compile-verified
//
#include <hip/hip_runtime.h>

typedef float v2f __attribute__((ext_vector_type(2)));
typedef float v8f __attribute__((ext_vector_type(8)));

#define NBIN   25
#define KDIM   128
#define BATCH  16
// ws layout (floats): [0,400) hLab[16][25], [400,800) hAll[16][25],
//                     [800,816) labsum[16], [816,832) qn[16]
#define WS_HALL 400
#define WS_LSUM 800
#define WS_QN   816
#define WS_TOT  832

__global__ void qap_init(const float* __restrict__ qX, float* __restrict__ ws) {
  int tid = threadIdx.x;
  for (int i = tid; i < WS_QN; i += blockDim.x) ws[i] = 0.0f;
  if (tid < BATCH) {
    const float* q = qX + tid * KDIM;
    float s = 0.0f;
    #pragma unroll
    for (int k = 0; k < KDIM; ++k) { float v = q[k]; s = fmaf(v, v, s); }
    ws[WS_QN + tid] = sqrtf(s);
  }
}

__global__ void __launch_bounds__(256)
qap_main(const float* __restrict__ qX, const float* __restrict__ dXs,
         const int* __restrict__ labels, float* __restrict__ ws, int D) {
  const int b    = blockIdx.y;
  const int tid  = threadIdx.x;
  const int lane = tid & 31;
  const int wave = tid >> 5;
  const int hi   = lane >> 4;   // 0 or 1 (K-pair select)
  const int lo   = lane & 15;   // matrix row within tile

  __shared__ float sq[KDIM];
  __shared__ float sdots[8][16];
  __shared__ float hAll[NBIN];
  __shared__ float hLab[NBIN];
  __shared__ float lsum;

  if (tid < KDIM) sq[tid] = qX[b * KDIM + tid];
  for (int i = tid; i < NBIN; i += blockDim.x) { hAll[i] = 0.0f; hLab[i] = 0.0f; }
  if (tid == 0) lsum = 0.0f;
  __syncthreads();

  const float qn = ws[WS_QN + b];
  const int   T  = D >> 4;                       // 16-row tiles (D % 16 == 0)
  const int   wavesPerBatch = gridDim.x * (blockDim.x >> 5);
  const int   gw = blockIdx.x * (blockDim.x >> 5) + wave;

  for (int tile = gw; tile < T; tile += wavesPerBatch) {
    const int d0 = tile << 4;
    // A fragment source: row = d0+lo, columns k0 + 2*hi + {0,1}
    const float* rowp = dXs + ((size_t)b * D + d0 + lo) * KDIM + 2 * hi;
    // prefetch this wave's next tile (speculative; OOB prefetch is dropped)
    __builtin_prefetch(rowp + (size_t)wavesPerBatch * 16 * KDIM, 0, 1);

    v8f   acc = {0.f, 0.f, 0.f, 0.f, 0.f, 0.f, 0.f, 0.f};
    float ns  = 0.0f;
    #pragma unroll
    for (int k0 = 0; k0 < KDIM; k0 += 4) {
      v2f av = *(const v2f*)(rowp + k0);          // A: 16x4 f32 fragment
      v2f bv = *(const v2f*)(&sq[k0 + 2 * hi]);   // B: 4x16, q broadcast over N
      ns = fmaf(av.x, av.x, ns);
      ns = fmaf(av.y, av.y, ns);
      acc = __builtin_amdgcn_wmma_f32_16x16x4_f32(
          false, av, false, bv, (short)0, acc, false, false);
    }
    // D[i,j] = dot(row i) for every column j. VGPR r holds M = r + 8*hi.
    if (lo == 0) {
      #pragma unroll
      for (int r = 0; r < 8; ++r) sdots[wave][8 * hi + r] = acc[r];
    }
    __builtin_amdgcn_wave_barrier();
    asm volatile("" ::: "memory");

    // complete per-row norm^2: lane L holds half of row lo, partner L^16 the rest
    float ntot = ns + __shfl_xor(ns, 16);
    float dotv = sdots[wave][lo];

    if (lane < 16) {
      float dn  = sqrtf(ntot);
      float sim = dotv / fmaxf(qn * dn, 1e-8f);
      int   lab = labels[(size_t)b * D + d0 + lo];
      // triangular soft-assign: m = (1-sim)/DELTA; touches bins floor(m), floor(m)+1
      float m = (1.0f - sim) * 12.0f;             // 1/DELTA = (NBIN-1)/2 = 12
      m = fminf(fmaxf(m, 0.0f), 24.0f);
      int i0 = (int)m; if (i0 > 23) i0 = 23;
      float f  = m - (float)i0;
      float w0 = 1.0f - f;
      atomicAdd(&hAll[i0], w0);
      atomicAdd(&hAll[i0 + 1], f);
      if (lab != 0) {
        atomicAdd(&hLab[i0], w0);
        atomicAdd(&hLab[i0 + 1], f);
        atomicAdd(&lsum, 1.0f);
      }
    }
  }

  __syncthreads();
  for (int i = tid; i < NBIN; i += blockDim.x) {
    atomicAdd(&ws[b * NBIN + i], hLab[i]);
    atomicAdd(&ws[WS_HALL + b * NBIN + i], hAll[i]);
  }
  if (tid == 0) atomicAdd(&ws[WS_LSUM + b], lsum);
}

__global__ void qap_final(const float* __restrict__ ws, float* __restrict__ out) {
  __shared__ float apq[BATCH];
  int t = threadIdx.x;
  if (t < BATCH) {
    float ls = ws[WS_LSUM + t];
    float cl = 0.0f, ca = 0.0f, s = 0.0f;
    #pragma unroll
    for (int n = 0; n < NBIN; ++n) {
      float hl = ws[t * NBIN + n];
      float ha = ws[WS_HALL + t * NBIN + n];
      cl += hl;
      ca += ha;
      float prec = cl / (ca + 1e-16f);
      float rec  = hl / ls;
      s += prec * rec;
    }
    apq[t] = s * (1.0f / NBIN);
  }
  __syncthreads();
  if (t == 0) {
    float tot = 0.0f;
    for (int i = 0; i < BATCH; ++i) tot += apq[i];
    out[0] = tot * (1.0f / BATCH);
  }
}

extern "C" void kernel_launch(void* const* d_in, const int* in_sizes, int n_in,
                              void* d_out, int out_size, void* d_ws, size_t ws_size,
                              hipStream_t stream) {
  const float* qX     = (const float*)d_in[0];
  const float* dXs    = (const float*)d_in[1];
  const int*   labels = (const int*)d_in[2];
  float*       out    = (float*)d_out;
  float*       ws     = (float*)d_ws;
  const int    D      = in_sizes[2] / BATCH;   // 100000

  qap_init<<<dim3(1), dim3(256), 0, stream>>>(qX, ws);
  qap_main<<<dim3(64, BATCH), dim3(256), 0, stream>>>(qX, dXs, labels, ws, D);
  qap_final<<<dim3(1), dim3(32), 0, stream>>>(ws, out);
}